// LiquidTimeConstantNetwork_35665408426494
// MI455X (gfx1250) — compile-verified
//
#include <hip/hip_runtime.h>
#include <math.h>

typedef __attribute__((ext_vector_type(2))) float v2f;
typedef __attribute__((ext_vector_type(8))) float v8f;

#define B_ 32
#define T_ 1024
#define I_ 256
#define H_ 512
#define O_ 256
#define DT_ 0.1f

__device__ __forceinline__ v2f ldg2(const float* p) { return *(const v2f*)p; }

// ---------------------------------------------------------------------------
// Init: zero hidden double-buffer and per-step barrier counters in d_ws.
// ---------------------------------------------------------------------------
__global__ __launch_bounds__(256) void ltc_init(int* __restrict__ p, int n) {
  int i = blockIdx.x * blockDim.x + threadIdx.x;
  if (i < n) p[i] = 0;
}

// ---------------------------------------------------------------------------
// 4-way-interleaved K loop: four independent WMMA accumulator chains so
// consecutive v_wmma_f32_16x16x4_f32 have no D->C RAW hazard (ISA 7.12.1),
// then elementwise-reduce (valid: matrix accumulate is elementwise over K).
// ---------------------------------------------------------------------------
__device__ __forceinline__ v8f wmma_dot_f32(const float* __restrict__ arow,
                                            const float* __restrict__ brow,
                                            int koff, int K) {
  v8f a0 = {}, a1 = {}, a2 = {}, a3 = {};
#pragma unroll 2
  for (int k0 = 0; k0 < K; k0 += 16) {
    v2f aa0 = ldg2(arow + k0 + 0 + koff);
    v2f bb0 = *(const v2f*)(brow + k0 + 0 + koff);
    v2f aa1 = ldg2(arow + k0 + 4 + koff);
    v2f bb1 = *(const v2f*)(brow + k0 + 4 + koff);
    v2f aa2 = ldg2(arow + k0 + 8 + koff);
    v2f bb2 = *(const v2f*)(brow + k0 + 8 + koff);
    v2f aa3 = ldg2(arow + k0 + 12 + koff);
    v2f bb3 = *(const v2f*)(brow + k0 + 12 + koff);
    a0 = __builtin_amdgcn_wmma_f32_16x16x4_f32(false, aa0, false, bb0,
                                               (short)0, a0, false, false);
    a1 = __builtin_amdgcn_wmma_f32_16x16x4_f32(false, aa1, false, bb1,
                                               (short)0, a1, false, false);
    a2 = __builtin_amdgcn_wmma_f32_16x16x4_f32(false, aa2, false, bb2,
                                               (short)0, a2, false, false);
    a3 = __builtin_amdgcn_wmma_f32_16x16x4_f32(false, aa3, false, bb3,
                                               (short)0, a3, false, false);
  }
  return (a0 + a1) + (a2 + a3);
}

// ---------------------------------------------------------------------------
// Phase 1: sensory[t][b][h] = sum_i x[b][t][i] * W_in[h][i]
// GEMM (T*B=32768 x H=512, K=256) with fp32 WMMA 16x16x4.
// One wave per 16x16 C tile; M index m = t*32 + b (row-gather from x).
// ---------------------------------------------------------------------------
__global__ __launch_bounds__(256) void ltc_proj(const float* __restrict__ x,
                                                const float* __restrict__ w_in,
                                                float* __restrict__ sens) {
  const int lane = threadIdx.x & 31;
  const int wave = threadIdx.x >> 5;
  const int tile = blockIdx.x * 8 + wave;   // 65536 tiles total
  const int mtile = tile >> 5;              // 2048 M tiles
  const int ntile = tile & 31;              // 32 N tiles
  const int half = lane >> 4;
  const int lm = lane & 15;

  // A fragment: lane holds row m, K pair at k0 + 2*half (ISA 16x4 f32 layout)
  const int m = mtile * 16 + lm;
  const int t = m >> 5, b = m & 31;
  const float* arow = x + ((size_t)b * T_ + t) * I_;
  // B fragment: lane holds column n (row of W_in), same K mapping
  const int n = ntile * 16 + lm;
  const float* brow = w_in + (size_t)n * I_;
  const int koff = half * 2;

  v8f acc = wmma_dot_f32(arow, brow, koff, I_);

  // C layout: lane -> column lm, VGPR v -> row v + 8*half
#pragma unroll
  for (int v = 0; v < 8; ++v) {
    int mg = mtile * 16 + v + 8 * half;
    int tt = mg >> 5, bb2 = mg & 31;
    sens[((size_t)tt * B_ + bb2) * H_ + n] = acc[v];
  }
}

// ---------------------------------------------------------------------------
// Phase 2: decay[t][h] = 1 - DT / softplus(tc[h] + 0.1*mean_b |sens[t,b,h]|)
// ---------------------------------------------------------------------------
__global__ __launch_bounds__(256) void ltc_decay(const float* __restrict__ sens,
                                                 const float* __restrict__ tc,
                                                 float* __restrict__ decay) {
  int idx = blockIdx.x * blockDim.x + threadIdx.x;  // t*H + h
  if (idx >= T_ * H_) return;
  int t = idx / H_, h = idx - t * H_;
  const float* p = sens + (size_t)t * B_ * H_ + h;
  float s = 0.f;
#pragma unroll 4
  for (int b = 0; b < B_; ++b) s += fabsf(p[b * H_]);
  float z = tc[h] + 0.1f * (s * (1.0f / B_));
  float tau = (z > 20.f) ? z : log1pf(expf(z));
  decay[idx] = 1.0f - DT_ / tau;
}

// ---------------------------------------------------------------------------
// Phase 3: the sequential scan. 48 persistent WGs (64 thr = 2 waves each).
//   WG 0..31 : hidden column tile n0 = wg*16      (W_rec slice in LDS)
//   WG 32..47: output column tile o0 = (wg-32)*16 (W_out slice in LDS)
// Hidden state double-buffered in global; per-step release/acquire counters.
// ---------------------------------------------------------------------------
__global__ __launch_bounds__(64) void ltc_scan(const float* __restrict__ sens,
                                               const float* __restrict__ decay,
                                               const float* __restrict__ w_rec,
                                               const float* __restrict__ w_out,
                                               const float* __restrict__ bias,
                                               float* __restrict__ out,   // (B,T,O) ++ (B,H)
                                               float* __restrict__ hbuf,  // 2*B*H
                                               int* __restrict__ cnt,     // T (hidden done)
                                               int* __restrict__ cnt2) {  // T (out done)
  __shared__ float lw[16 * 514];  // 16 rows x 512, stride 514 to spread banks
  const int wg = blockIdx.x;
  const int tid = threadIdx.x;
  const int lane = tid & 31;
  const int w = tid >> 5;         // wave id -> M tile (batch rows 16w..16w+15)
  const int half = lane >> 4;
  const int lm = lane & 15;
  const bool isH = (wg < 32);
  const int n0 = isH ? wg * 16 : (wg - 32) * 16;
  const float* W = isH ? w_rec : w_out;

  for (int i = tid; i < 16 * H_; i += 64) {
    int r = i >> 9, c = i & 511;
    lw[r * 514 + c] = W[(size_t)(n0 + r) * H_ + c];
  }
  __syncthreads();

  const int koff = half * 2;
  const float* brow = &lw[lm * 514];
  float* h0 = hbuf;
  float* h1 = hbuf + B_ * H_;

  for (int t = 0; t < T_; ++t) {
    const float* hcur = (t & 1) ? h1 : h0;
    float* hnxt = (t & 1) ? h0 : h1;
    if (isH) {
      if (tid == 0) {
        if (t >= 1)  // full hidden of step t-1 visible
          while (__hip_atomic_load(&cnt[t - 1], __ATOMIC_ACQUIRE,
                                   __HIP_MEMORY_SCOPE_AGENT) < 32)
            __builtin_amdgcn_s_sleep(1);
        if (t >= 2)  // out-WGs done reading buffer we are about to overwrite
          while (__hip_atomic_load(&cnt2[t - 2], __ATOMIC_ACQUIRE,
                                   __HIP_MEMORY_SCOPE_AGENT) < 16)
            __builtin_amdgcn_s_sleep(1);
      }
      __syncthreads();

      const float* arow = hcur + (size_t)(w * 16 + lm) * H_;
      v8f acc = wmma_dot_f32(arow, brow, koff, H_);

      const int n = n0 + lm;
      const float bn = bias[n];
      const float dc = decay[t * H_ + n];
#pragma unroll
      for (int v = 0; v < 8; ++v) {
        int b = w * 16 + v + 8 * half;
        size_t hi = (size_t)b * H_ + n;
        float hold = hcur[hi];
        float s = sens[((size_t)t * B_ + b) * H_ + n];
        float hnew = hold * dc + DT_ * tanhf(s + acc[v] + bn);
        hnxt[hi] = hnew;
        if (t == T_ - 1) out[(size_t)B_ * T_ * O_ + hi] = hnew;  // final hidden
      }
      __threadfence();
      __syncthreads();
      if (tid == 0)
        __hip_atomic_fetch_add(&cnt[t], 1, __ATOMIC_RELEASE,
                               __HIP_MEMORY_SCOPE_AGENT);
    } else {
      if (tid == 0)
        while (__hip_atomic_load(&cnt[t], __ATOMIC_ACQUIRE,
                                 __HIP_MEMORY_SCOPE_AGENT) < 32)
          __builtin_amdgcn_s_sleep(1);
      __syncthreads();

      const float* arow = hnxt + (size_t)(w * 16 + lm) * H_;  // new hidden h_{t+1}
      v8f acc = wmma_dot_f32(arow, brow, koff, H_);

      const int o = n0 + lm;
#pragma unroll
      for (int v = 0; v < 8; ++v) {
        int b = w * 16 + v + 8 * half;
        out[((size_t)b * T_ + t) * O_ + o] = acc[v];
      }
      __threadfence();
      __syncthreads();
      if (tid == 0)
        __hip_atomic_fetch_add(&cnt2[t], 1, __ATOMIC_RELEASE,
                               __HIP_MEMORY_SCOPE_AGENT);
    }
  }
}

// ---------------------------------------------------------------------------
extern "C" void kernel_launch(void* const* d_in, const int* in_sizes, int n_in,
                              void* d_out, int out_size, void* d_ws, size_t ws_size,
                              hipStream_t stream) {
  const float* x     = (const float*)d_in[0];  // (B,T,I)
  const float* w_in  = (const float*)d_in[1];  // (H,I)
  const float* w_rec = (const float*)d_in[2];  // (H,H)
  const float* tc    = (const float*)d_in[3];  // (H,)
  const float* w_out = (const float*)d_in[4];  // (O,H)
  const float* bias  = (const float*)d_in[5];  // (H,)
  float* out = (float*)d_out;

  // workspace layout (floats/ints, all 4B aligned)
  float* sens  = (float*)d_ws;                         // T*B*H
  float* decay = sens + (size_t)T_ * B_ * H_;          // T*H
  float* hbuf  = decay + (size_t)T_ * H_;              // 2*B*H
  int*   cnt   = (int*)(hbuf + 2 * B_ * H_);           // T
  int*   cnt2  = cnt + T_;                             // T

  // zero hidden buffers + counters (floats zero-bits == int 0)
  int nz = 2 * B_ * H_ + 2 * T_;
  ltc_init<<<(nz + 255) / 256, 256, 0, stream>>>((int*)hbuf, nz);

  // Phase 1: input projection (65536 tiles, 8 waves/block)
  ltc_proj<<<8192, 256, 0, stream>>>(x, w_in, sens);

  // Phase 2: adaptive time-constant decay factors
  ltc_decay<<<(T_ * H_ + 255) / 256, 256, 0, stream>>>(sens, tc, decay);

  // Phase 3: sequential scan with 48 persistent workgroups
  ltc_scan<<<48, 64, 0, stream>>>(sens, decay, w_rec, w_out, bias,
                                  out, hbuf, cnt, cnt2);
}